// BottomLevel_56427280335297
// MI455X (gfx1250) — compile-verified
//
#include <hip/hip_runtime.h>

// CDNA5 (gfx1250) correlation / cost-volume kernel using fp32 WMMA.
// out[b,h,w,dy,dx] = sum_c p[b,h,w,c] * q[b, h+2+dy, w+2+dx, c]
//
// Band-GEMM mapping: per 16-wide pixel tile and per dy, two chained
// V_WMMA_F32_16X16X4_F32 accumulations over K=C=128 produce 16x16 tiles whose
// diagonals n-m in [0,8] are the 9 dx outputs.

typedef float v2f __attribute__((ext_vector_type(2)));
typedef float v8f __attribute__((ext_vector_type(8)));

#define WIN   9
#define OFF   2
#define CCH   128
#define BB    4
#define HH    512
#define WW    512
#define HO    502
#define WO    502

__global__ __launch_bounds__(256) void corr_wmma_f32_kernel(
    const float* __restrict__ p,
    const float* __restrict__ q,
    float* __restrict__ out)
{
    const int lane = threadIdx.x & 31;
    const int wave = threadIdx.x >> 5;
    const int w0s  = blockIdx.x * 128 + wave * 16;   // this wave's 16-wide pixel subtile
    const int h    = blockIdx.y;
    const int b    = blockIdx.z;

    const int nIdx = lane & 15;   // N (and A-row M) index within lane-half
    const int half = lane >> 4;   // 0: holds K={0,1}; 1: holds K={2,3}  (ISA 16x4 f32 A/B layout)

    // ---------- A operand: p tile (16 x 128), loaded once, reused for all 81 taps ----------
    const int prow = min(w0s + nIdx, WO - 1);        // clamp; out-of-range rows never stored
    const float* pbase = p + (((size_t)b * HO + h) * WO + prow) * CCH + half * 2;
    v2f a[32];
#pragma unroll
    for (int k = 0; k < 32; ++k)
        a[k] = *(const v2f*)(pbase + 4 * k);         // global_load_b64, 8B aligned

    const size_t outRowBase = (((size_t)b * HO + h) * WO) * (size_t)(WIN * WIN);

    for (int dy = 0; dy < WIN; ++dy) {
        const int qrow = h + OFF + dy;               // always < 512 since h <= 501, dy <= 8
        // B columns: q positions [w0s+2 .. w0s+33] in two groups of 16.
        const int col0 = min(w0s + OFF + nIdx,      WW - 1);
        const int col1 = min(w0s + OFF + 16 + nIdx, WW - 1);  // clamp only pads unused band
        const float* qb0 = q + (((size_t)b * HH + qrow) * WW + col0) * CCH + half * 2;
        const float* qb1 = q + (((size_t)b * HH + qrow) * WW + col1) * CCH + half * 2;

        // Prefetch next dy's q row while this row's WMMAs run (global_prefetch_b8).
        if (dy + 1 < WIN) {
            __builtin_prefetch(qb0 + (size_t)WW * CCH, 0, 0);
            __builtin_prefetch(qb1 + (size_t)WW * CCH, 0, 0);
        }

        v8f acc0 = {};
        v8f acc1 = {};
#pragma unroll
        for (int k = 0; k < 32; ++k) {
            v2f b0 = *(const v2f*)(qb0 + 4 * k);
            v2f b1 = *(const v2f*)(qb1 + 4 * k);
            // 8 args: (neg_a, A, neg_b, B, c_mod, C, reuse_a, reuse_b)
            acc0 = __builtin_amdgcn_wmma_f32_16x16x4_f32(
                       false, a[k], false, b0, (short)0, acc0, false, false);
            acc1 = __builtin_amdgcn_wmma_f32_16x16x4_f32(
                       false, a[k], false, b1, (short)0, acc1, false, false);
        }

        // ---------- scatter the 9 valid diagonals (band) of the two 16x16 tiles ----------
        // C/D layout: VGPR i, lanes 0-15 -> (M=i,   N=lane);
        //                      lanes 16-31 -> (M=i+8, N=lane-16)
        const int n  = nIdx;
        const int mo = half * 8;
#pragma unroll
        for (int i = 0; i < 8; ++i) {
            const int m = i + mo;
            const int w = w0s + m;
            if (w < WO) {
                const size_t ob = outRowBase + (size_t)w * (WIN * WIN) + dy * WIN;
                const int dx0 = n - m;            // group 0: q col = w0s+2+n
                if ((unsigned)dx0 < WIN)
                    out[ob + dx0] = acc0[i];
                const int dx1 = n + 16 - m;       // group 1: q col = w0s+18+n
                if ((unsigned)dx1 < WIN)
                    out[ob + dx1] = acc1[i];
            }
        }
    }
}

extern "C" void kernel_launch(void* const* d_in, const int* in_sizes, int n_in,
                              void* d_out, int out_size, void* d_ws, size_t ws_size,
                              hipStream_t stream) {
    (void)in_sizes; (void)n_in; (void)d_ws; (void)ws_size; (void)out_size;
    const float* p = (const float*)d_in[0];   // (4, 502, 502, 128) f32
    const float* q = (const float*)d_in[1];   // (4, 512, 512, 128) f32
    float* out = (float*)d_out;               // (4, 502, 502, 9, 9) f32

    dim3 grid((WO + 127) / 128, HO, BB);      // (4, 502, 4)
    dim3 block(256);                          // 8 wave32s, one 16-wide subtile each
    corr_wmma_f32_kernel<<<grid, block, 0, stream>>>(p, q, out);
}